// CorrelationCalculator_48541720379860
// MI455X (gfx1250) — compile-verified
//
#include <hip/hip_runtime.h>

typedef __attribute__((ext_vector_type(2))) float v2f;
typedef __attribute__((ext_vector_type(8))) float v8f;

#define PATCH_S 15
#define PATCH   29
#define Bn 4
#define Cn 256
#define Hn 64
#define Wn 64
#define HW  (Hn * Wn)      // 4096
#define CHW (Cn * HW)      // 1048576

__global__ void zero_f4(float4* __restrict__ p, int n4) {
  int i = blockIdx.x * blockDim.x + threadIdx.x;
  float4 z; z.x = 0.f; z.y = 0.f; z.z = 0.f; z.w = 0.f;
  if (i < n4) p[i] = z;
}

__launch_bounds__(256)
__global__ void corr_wmma_f32(const float* __restrict__ f1,
                              const float* __restrict__ f2,
                              float* __restrict__ corr,
                              float* __restrict__ sim) {
  // A tile [m=16][k=256] with +4 dword pad per row: bank-conflict-free b64 reads
  __shared__ float aTile[16 * 260];    // 16640 B
  // per-wave 16x16 staging + dump slots (index 240..255 are write-only garbage)
  __shared__ float stage[8 * 256];

  const int tid  = threadIdx.x;
  const int lane = tid & 31;
  const int wid  = tid >> 5;
  const int j0   = blockIdx.x << 4;   // 16-wide pixel tile
  const int i    = blockIdx.y;
  const int b    = blockIdx.z;

  // ---- cooperative stage of A = f1[b, :, i, j0:j0+16] into LDS ----
  {
    const int k = tid;  // 256 threads == 256 channels
    const float4* s4 = reinterpret_cast<const float4*>(
        f1 + (size_t)b * CHW + (size_t)k * HW + i * Wn + j0);
#pragma unroll
    for (int q = 0; q < 4; ++q) {
      float4 v = s4[q];
      aTile[(4 * q + 0) * 260 + k] = v.x;
      aTile[(4 * q + 1) * 260 + k] = v.y;
      aTile[(4 * q + 2) * 260 + k] = v.z;
      aTile[(4 * q + 3) * 260 + k] = v.w;
    }
  }
  __syncthreads();

  // WMMA f32 16x16x4 lane roles (wave32):
  //  A: lanes 0-15 M=lane hold K=k0,k0+1 ; lanes 16-31 hold K=k0+2,k0+3
  //  B: lanes 0-15 N=lane hold K=k0,k0+1 ; lanes 16-31 hold K=k0+2,k0+3
  const int mn  = lane & 15;          // M (A) / N (B) index
  const int sel = (lane >> 4) << 1;   // K sub-select: 0 or 2
  const int mhi = (lane >> 4) << 3;   // D-matrix M offset: 0 or 8

  float* st = stage + wid * 256;

  for (int pi = wid; pi < PATCH_S; pi += 8) {
    const int  i2    = i + pi - 7;
    const bool rowOK = (i2 >= 0) && (i2 < Hn);
    const int  i2c   = i2 < 0 ? 0 : (i2 > Hn - 1 ? Hn - 1 : i2);

    // B tiles cover f2 columns [j0-7 .. j0+8] and [j0+9 .. j0+24] (clamped)
    int c0 = j0 - 7 + mn; c0 = c0 < 0 ? 0 : (c0 > Wn - 1 ? Wn - 1 : c0);
    int c1 = j0 + 9 + mn; c1 = c1 < 0 ? 0 : (c1 > Wn - 1 ? Wn - 1 : c1);

    const float* p2 = f2 + (size_t)b * CHW + i2c * Wn;
    const float* lA = aTile + mn * 260 + sel;

    v8f acc0 = {0.f, 0.f, 0.f, 0.f, 0.f, 0.f, 0.f, 0.f};
    v8f acc1 = {0.f, 0.f, 0.f, 0.f, 0.f, 0.f, 0.f, 0.f};

#pragma unroll 4
    for (int k0 = 0; k0 < Cn; k0 += 4) {
      v2f a = *reinterpret_cast<const v2f*>(lA + k0);   // ds_load_b64
      const float* pk = p2 + (size_t)(k0 + sel) * HW;
      v2f b0, b1;
      b0.x = pk[c0];      b0.y = pk[HW + c0];
      b1.x = pk[c1];      b1.y = pk[HW + c1];
      acc0 = __builtin_amdgcn_wmma_f32_16x16x4_f32(false, a, false, b0,
                                                   (short)0, acc0, false, false);
      acc1 = __builtin_amdgcn_wmma_f32_16x16x4_f32(false, a, false, b1,
                                                   (short)0, acc1, false, false);
    }

    // ---- extract diagonals: sim(pj, m) = gram[m][m + pj] ----
    // Branch-free: invalid (lane,v,t) elements are steered to dump slot 240.
#pragma unroll
    for (int t = 0; t < 2; ++t) {
      const int  g     = mn + (t << 4);        // gram column 0..31
      const int  f2col = j0 - 7 + g;           // absolute f2 column
      const bool ok    = rowOK && (f2col >= 0) && (f2col < Wn);
      v8f acc = t ? acc1 : acc0;
#pragma unroll
      for (int v = 0; v < 8; ++v) {
        const int  m_out = v + mhi;            // output pixel m = 0..15
        const int  pj    = g - m_out;          // displacement column
        const bool pjOK  = (pj >= 0) && (pj < PATCH_S);
        const int  idx   = pjOK ? (pj * 16 + m_out) : 240;  // 240 = dump
        st[idx] = (ok && pjOK) ? acc[v] : 0.f;
      }
    }
    asm volatile("s_wait_dscnt 0" ::: "memory");

    // ---- coalesced stores: 2 pj-rows x 16 contiguous pixels per pass ----
    // r = 0..6 covers pj 0..13 (both halves valid); pj=14 done by both halves
    // redundantly (same address, same value) to stay branch-free.
    const int half = lane >> 4;
    const int simBase  = (((b * PATCH_S + pi) * PATCH_S) * Hn + i) * Wn + j0 + mn;
    const int corrBase = (((b * PATCH + pi + 7) * PATCH + 7) * Hn + i) * Wn + j0 + mn;
#pragma unroll
    for (int r = 0; r < 7; ++r) {
      const int pj = (r << 1) + half;
      const float val = st[pj * 16 + mn];
      sim[simBase + pj * HW]   = val;
      corr[corrBase + pj * HW] = val;
    }
    {
      const float val = st[14 * 16 + mn];       // both halves: same addr/value
      sim[simBase + 14 * HW]   = val;
      corr[corrBase + 14 * HW] = val;
    }
  }
}

extern "C" void kernel_launch(void* const* d_in, const int* in_sizes, int n_in,
                              void* d_out, int out_size, void* d_ws, size_t ws_size,
                              hipStream_t stream) {
  const float* f1 = (const float*)d_in[0];
  const float* f2 = (const float*)d_in[1];
  float* out  = (float*)d_out;
  float* corr = out;                                       // [4,29,29,64,64]
  float* sim  = out + (size_t)Bn * PATCH * PATCH * HW;     // [4,15,15,64,64]

  // zero-fill both outputs (pad borders must be deterministic zeros)
  int n4 = out_size >> 2;  // out_size divisible by 4
  zero_f4<<<dim3((n4 + 255) / 256), dim3(256), 0, stream>>>((float4*)d_out, n4);

  dim3 grid(Wn / 16, Hn, Bn);   // 4 x 64 x 4 = 1024 workgroups
  corr_wmma_f32<<<grid, dim3(256), 0, stream>>>(f1, f2, corr, sim);
}